// HypAgg_50268297232886
// MI455X (gfx1250) — compile-verified
//
#include <hip/hip_runtime.h>
#include <cstdint>
#include <cstddef>

#define DIM 128
#define WAVES_PER_BLOCK 8
#define SLIST_CAP 64

// ---------------------------------------------------------------------------
// wave32 all-reduce (gfx1250 is wave32-only)
// ---------------------------------------------------------------------------
__device__ __forceinline__ float wave_allreduce_sum(float v) {
    v += __shfl_xor(v, 16, 32);
    v += __shfl_xor(v, 8, 32);
    v += __shfl_xor(v, 4, 32);
    v += __shfl_xor(v, 2, 32);
    v += __shfl_xor(v, 1, 32);
    return v;
}

// ---------------------------------------------------------------------------
// K1: zero the degree histogram
// ---------------------------------------------------------------------------
__global__ void k_zero(int* __restrict__ p, int n) {
    int i = blockIdx.x * blockDim.x + threadIdx.x;
    if (i < n) p[i] = 0;
}

// ---------------------------------------------------------------------------
// K2: degree histogram over destination nodes (int atomics only)
// ---------------------------------------------------------------------------
__global__ void k_hist(const int* __restrict__ dst, int E, int* __restrict__ counts) {
    int e = blockIdx.x * blockDim.x + threadIdx.x;
    if (e < E) atomicAdd(&counts[dst[e]], 1);
}

// ---------------------------------------------------------------------------
// K3: single-block exclusive prefix scan of counts -> offsets (and cursor copy)
// ---------------------------------------------------------------------------
__global__ __launch_bounds__(1024) void k_scan(const int* __restrict__ counts, int N,
                                               int* __restrict__ offsets,
                                               int* __restrict__ cursor) {
    __shared__ int part[1024];
    const int t = threadIdx.x;
    const int chunk = (N + 1023) >> 10;
    const int base = t * chunk;
    const int end = min(base + chunk, N);
    int s = 0;
    for (int i = base; i < end; ++i) s += counts[i];
    part[t] = s;
    __syncthreads();
    for (int d = 1; d < 1024; d <<= 1) {
        int v = (t >= d) ? part[t - d] : 0;
        __syncthreads();
        part[t] += v;
        __syncthreads();
    }
    int run = (t == 0) ? 0 : part[t - 1];
    for (int i = base; i < end; ++i) {
        offsets[i] = run;
        cursor[i] = run;
        run += counts[i];
    }
}

// ---------------------------------------------------------------------------
// K4: scatter source indices into CSR buckets
// ---------------------------------------------------------------------------
__global__ void k_fill(const int* __restrict__ dst, const int* __restrict__ src, int E,
                       int* __restrict__ cursor, int* __restrict__ csr_src) {
    int e = blockIdx.x * blockDim.x + threadIdx.x;
    if (e < E) {
        int pos = atomicAdd(&cursor[dst[e]], 1);
        csr_src[pos] = src[e];
    }
}

// ---------------------------------------------------------------------------
// K5: fused logmap0-gather-segment_sum-expmap0-proj.
// One wave32 per destination row; each lane owns 4 of the 128 columns.
// Neighbor rows (512 B each) are DMA'd global->LDS with the gfx1250 async
// path (global_load_async_to_lds_b128 / s_wait_asynccnt), double-buffered.
// ---------------------------------------------------------------------------
__global__ __launch_bounds__(256) void k_agg(const float* __restrict__ x,
                                             const int* __restrict__ offsets,
                                             const int* __restrict__ counts,
                                             const int* __restrict__ csr_src,
                                             float* __restrict__ out, int N) {
    __shared__ __align__(16) float gbuf[WAVES_PER_BLOCK][2][DIM];  // 8 KB: double buffer / wave
    __shared__ int slist[WAVES_PER_BLOCK][SLIST_CAP];              // 2 KB: sorted neighbor ids

    const int lane = threadIdx.x & 31;
    const int wave = threadIdx.x >> 5;
    const int row  = blockIdx.x * WAVES_PER_BLOCK + wave;
    if (row >= N) return;  // row is wave-uniform: whole wave exits together

    const int start = offsets[row];
    const int deg   = counts[row];
    const int m     = (deg < SLIST_CAP) ? deg : SLIST_CAP;

    // Load this row's neighbor list into LDS and sort it (lane 0, insertion
    // sort; avg degree ~6) so the FP accumulation order is deterministic
    // regardless of the atomic fill order in k_fill.
    for (int j = lane; j < m; j += 32) slist[wave][j] = csr_src[start + j];
    __builtin_amdgcn_wave_barrier();
    if (lane == 0) {
        for (int i = 1; i < m; ++i) {
            int key = slist[wave][i];
            int k = i - 1;
            while (k >= 0 && slist[wave][k] > key) { slist[wave][k + 1] = slist[wave][k]; --k; }
            slist[wave][k + 1] = key;
        }
    }
    __builtin_amdgcn_wave_barrier();

    // Per-lane LDS byte address (generic LDS pointer low 32 bits == LDS offset).
    const unsigned lds_base =
        ((unsigned)(uintptr_t)&gbuf[wave][0][0]) + (unsigned)lane * 16u;

    auto neighbor = [&](int j) -> int {
        return (j < SLIST_CAP) ? slist[wave][j] : csr_src[start + j];
    };
    // Async DMA one 512B row into LDS: each lane copies its 16B (b128).
    auto issue = [&](int j, int buf) {
        const int src_row = neighbor(j);
        const unsigned long long ga =
            (unsigned long long)(uintptr_t)(x + (size_t)src_row * DIM + lane * 4);
        const unsigned la = lds_base + (unsigned)buf * (DIM * 4u);
        asm volatile("global_load_async_to_lds_b128 %0, %1, off"
                     :: "v"(la), "v"(ga) : "memory");
    };

    float4 acc = make_float4(0.f, 0.f, 0.f, 0.f);
    if (deg > 0) issue(0, 0);
    for (int j = 0; j < deg; ++j) {
        const int buf = j & 1;
        if (j + 1 < deg) {
            issue(j + 1, buf ^ 1);
            asm volatile("s_wait_asynccnt 1" ::: "memory");  // slot j landed (in-order)
        } else {
            asm volatile("s_wait_asynccnt 0" ::: "memory");
        }
        const float4 v = *(const float4*)&gbuf[wave][buf][lane * 4];

        // logmap0: scale = artanh(min(||p||, 1-1e-7)) / max(||p||, 1e-15)
        float ss = v.x * v.x + v.y * v.y + v.z * v.z + v.w * v.w;
        ss = wave_allreduce_sum(ss);
        const float pn  = fmaxf(sqrtf(ss), 1e-15f);
        const float tcl = fminf(pn, 1.0f - 1e-7f);
        const float at  = 0.5f * (log1pf(tcl) - log1pf(-tcl));
        const float s   = at / pn;
        acc.x += v.x * s; acc.y += v.y * s; acc.z += v.z * s; acc.w += v.w * s;
    }

    // expmap0 + proj: ||expmap0(u)|| = tanh(||u||), so proj folds into one scale.
    float ss = acc.x * acc.x + acc.y * acc.y + acc.z * acc.z + acc.w * acc.w;
    ss = wave_allreduce_sum(ss);
    const float un = fmaxf(sqrtf(ss), 1e-15f);
    const float th = tanhf(un);
    const float maxnorm = 1.0f - 4e-3f;
    const float s = (th > maxnorm) ? (maxnorm / un) : (th / un);
    float4 y = make_float4(acc.x * s, acc.y * s, acc.z * s, acc.w * s);
    *(float4*)(out + (size_t)row * DIM + lane * 4) = y;
}

// ---------------------------------------------------------------------------
// Host launcher. Workspace layout (ints): counts[N] | offsets[N] | cursor[N] |
// csr_src[E]  -> (3N + E)*4 = 3.6 MB for N=100000, E=600000.
// ---------------------------------------------------------------------------
extern "C" void kernel_launch(void* const* d_in, const int* in_sizes, int n_in,
                              void* d_out, int out_size, void* d_ws, size_t ws_size,
                              hipStream_t stream) {
    const float* x   = (const float*)d_in[0];
    const int*   adj = (const int*)d_in[1];
    const int N = in_sizes[0] / DIM;   // 100000
    const int E = in_sizes[1] / 2;     // 600000
    const int* dst = adj;              // adj[0, :]
    const int* src = adj + E;          // adj[1, :]

    int* counts  = (int*)d_ws;
    int* offsets = counts + N;
    int* cursor  = offsets + N;
    int* csr_src = cursor + N;
    float* out = (float*)d_out;

    k_zero<<<(N + 255) / 256, 256, 0, stream>>>(counts, N);
    k_hist<<<(E + 255) / 256, 256, 0, stream>>>(dst, E, counts);
    k_scan<<<1, 1024, 0, stream>>>(counts, N, offsets, cursor);
    k_fill<<<(E + 255) / 256, 256, 0, stream>>>(dst, src, E, cursor, csr_src);
    k_agg<<<(N + WAVES_PER_BLOCK - 1) / WAVES_PER_BLOCK, 256, 0, stream>>>(
        x, offsets, counts, csr_src, out, N);
}